// ReferenceRhythmEncoder_4088808865914
// MI455X (gfx1250) — compile-verified
//
#include <hip/hip_runtime.h>
#include <hip/hip_bf16.h>

// Problem constants (match reference setup_inputs / module constants)
#define BB 64
#define TT 8192
#define FF 80
#define BINS 24          // TRACE_BINS
#define WIN 24           // window_size (python scalar, fixed at 24)
#define HORIZON_F 0.35f
#define NTHREADS 256
#define CHUNK (TT / NTHREADS)   // 32 mask elements per thread

typedef __attribute__((ext_vector_type(2))) float v2f;
typedef __attribute__((ext_vector_type(8))) float v8f;

__global__ __launch_bounds__(NTHREADS)
void rhythm_encoder_kernel(const float* __restrict__ feat,   // [B,T,F]
                           const int*   __restrict__ mask,   // [B,T]
                           const float* __restrict__ phase,  // [B]
                           float* __restrict__ out_sampled,  // [B,WIN,F]
                           float* __restrict__ out_trace,    // [B,BINS,F]
                           float* __restrict__ out_runmean)  // [B]
{
    const int b    = blockIdx.x;
    const int tid  = threadIdx.x;
    const int lane = tid & 31;
    const int wave = tid >> 5;

    __shared__ float s_csum[TT];          // 32 KB: inclusive cumsum (float)
    __shared__ int   s_part[NTHREADS];    // scan partials
    __shared__ int   s_rises;
    __shared__ float s_trace[BINS][FF];   // 24x80 trace tile
    __shared__ float s_W[32][BINS];       // 32x24 sampling matrix (rows 24..31 zero)
    __shared__ int   s_left[BINS], s_rightc[BINS], s_mode[BINS];
    __shared__ float s_alpha[BINS];

    const int*   mrow = mask + (size_t)b * TT;
    const float* frow = feat + (size_t)b * TT * FF;

    __builtin_prefetch(mrow, 0, 3);   // global_prefetch_b8

    if (tid == 0) s_rises = 0;
    int* s_ci = reinterpret_cast<int*>(s_csum);

    // ---- pass 1: stash mask into LDS, local sums + rising edges ----
    const int t0 = tid * CHUNK;
    int lsum = 0, lrise = 0;
    int prev = (t0 > 0) ? mrow[t0 - 1] : 0;
    for (int i = 0; i < CHUNK; ++i) {
        int m = mrow[t0 + i];
        s_ci[t0 + i] = m;
        lsum += m;
        lrise += (m == 1 && prev == 0) ? 1 : 0;
        prev = m;
    }
    s_part[tid] = lsum;
    if (lrise) atomicAdd(&s_rises, lrise);
    __syncthreads();

    // ---- inclusive scan of per-thread sums (Hillis-Steele over 256) ----
    for (int off = 1; off < NTHREADS; off <<= 1) {
        int v = (tid >= off) ? s_part[tid - off] : 0;
        __syncthreads();
        s_part[tid] += v;
        __syncthreads();
    }
    const int prefix  = s_part[tid] - lsum;        // exclusive prefix
    const int total_i = s_part[NTHREADS - 1];
    const int rises   = s_rises;

    // ---- pass 2: materialize float inclusive cumsum in LDS ----
    int running = prefix;
    for (int i = 0; i < CHUNK; ++i) {
        running += s_ci[t0 + i];
        s_csum[t0 + i] = (float)running;
    }
    __syncthreads();

    // ---- run_mean ----
    if (tid == 0) {
        float ones = (float)total_i;
        out_runmean[b] = (rises > 0) ? (ones / fmaxf((float)rises, 1.0f)) : 0.0f;
    }

    const bool  use_uniform = (total_i <= 0);
    const float ftotal      = (float)total_i;
    const float inv_tm1     = 1.0f / (float)(TT - 1);
    auto progressAt = [&](int t) -> float {
        return use_uniform ? ((float)t * inv_tm1) : (s_csum[t] / ftotal);
    };

    // ---- searchsorted(side=left) for 24 targets, by lanes 0..23 ----
    if (tid < BINS) {
        const float target = (float)tid / 23.0f;   // linspace(0,1,24)
        int lo = 0, hi = TT;
        while (lo < hi) {
            int mid = (lo + hi) >> 1;
            if (progressAt(mid) < target) lo = mid + 1; else hi = mid;
        }
        const int right = lo;
        const int left  = min(max(right - 1, 0), TT - 1);
        const int rc    = min(right, TT - 1);
        const float lp = progressAt(left);
        const float rp = progressAt(rc);
        const float denom = fmaxf(fabsf(rp - lp), 1e-6f);
        s_alpha[tid]  = fminf(fmaxf((target - lp) / denom, 0.0f), 1.0f);
        s_left[tid]   = left;
        s_rightc[tid] = rc;
        s_mode[tid]   = (right <= 0) ? 0 : ((right >= TT) ? 1 : 2);
        __builtin_prefetch(frow + (size_t)left * FF, 0, 3);
        __builtin_prefetch(frow + (size_t)rc   * FF, 0, 3);
    }
    __syncthreads();

    // ---- gather + lerp feature rows -> trace (LDS + global out) ----
    for (int idx = tid; idx < BINS * FF; idx += NTHREADS) {
        const int j = idx / FF, f = idx - j * FF;
        const int mode = s_mode[j];
        float v;
        if (mode == 0)      v = frow[f];
        else if (mode == 1) v = frow[(size_t)(TT - 1) * FF + f];
        else {
            const float a  = s_alpha[j];
            const float fl = frow[(size_t)s_left[j]   * FF + f];
            const float fr = frow[(size_t)s_rightc[j] * FF + f];
            v = fl * (1.0f - a) + fr * a;
        }
        s_trace[j][f] = v;
        out_trace[((size_t)b * BINS + j) * FF + f] = v;
    }

    // ---- build 32x24 sampling matrix W (2 non-zeros per real row) ----
    if (tid < 32) {
        for (int k = 0; k < BINS; ++k) s_W[tid][k] = 0.0f;
    }
    __syncthreads();
    if (tid < WIN) {
        const float ph  = phase[b];
        const float off = (float)tid * (HORIZON_F / (float)(WIN - 1));
        const float pos = fminf(fmaxf(ph + off, 0.0f), 1.0f);
        const float sc  = pos * (float)(BINS - 1);
        const int   l   = min(max((int)floorf(sc), 0), BINS - 1);
        const int   r   = min(l + 1, BINS - 1);
        const float a   = sc - (float)l;
        s_W[tid][l] += 1.0f - a;
        s_W[tid][r] += a;
    }
    __syncthreads();

    // ---- sampled = W(32x24) x trace(24x80) via V_WMMA_F32_16X16X4_F32 ----
    // 10 output tiles: tm in {0,1} x tn in {0..4}; 8 waves, K=24 in 6 steps of 4.
    // A frag (16x4 f32): lane = M (lane&15), VGPR v + 2*(lane>>4) = K.
    // B frag (4x16 f32): lane = N (lane&15), VGPR v + 2*(lane>>4) = K.
    const int half = lane >> 4;
    for (int job = wave; job < 10; job += (NTHREADS / 32)) {
        const int tm = job / 5, tn = job - tm * 5;
        const int mi  = tm * 16 + (lane & 15);
        const int col = tn * 16 + (lane & 15);
        v8f c = {};
        for (int ks = 0; ks < 6; ++ks) {
            const int ka = ks * 4 + half * 2;
            v2f a, bb;
            a.x  = s_W[mi][ka];
            a.y  = s_W[mi][ka + 1];
            bb.x = s_trace[ka][col];
            bb.y = s_trace[ka + 1][col];
            c = __builtin_amdgcn_wmma_f32_16x16x4_f32(
                    /*neg_a=*/false, a, /*neg_b=*/false, bb,
                    /*c_mod=*/(short)0, c, /*reuse_a=*/false, /*reuse_b=*/false);
        }
        for (int v = 0; v < 8; ++v) {
            const int row = tm * 16 + v + half * 8;
            if (row < WIN)
                out_sampled[((size_t)b * WIN + row) * FF + col] = c[v];
        }
    }
}

extern "C" void kernel_launch(void* const* d_in, const int* in_sizes, int n_in,
                              void* d_out, int out_size, void* d_ws, size_t ws_size,
                              hipStream_t stream) {
    (void)in_sizes; (void)n_in; (void)out_size; (void)d_ws; (void)ws_size;
    const float* feat  = (const float*)d_in[0];
    const int*   mask  = (const int*)d_in[1];
    const float* phase = (const float*)d_in[2];
    // d_in[3] = window_size scalar (== 24, baked in as WIN)

    float* out          = (float*)d_out;
    float* out_sampled  = out;                                   // B*WIN*F
    float* out_trace    = out_sampled + (size_t)BB * WIN * FF;   // B*BINS*F
    float* out_runmean  = out_trace   + (size_t)BB * BINS * FF;  // B

    hipLaunchKernelGGL(rhythm_encoder_kernel, dim3(BB), dim3(NTHREADS), 0, stream,
                       feat, mask, phase, out_sampled, out_trace, out_runmean);
}